// EdgeTypeGNNLayer_9517647528114
// MI455X (gfx1250) — compile-verified
//
#include <hip/hip_runtime.h>

#define IN_DIM   128
#define OUT_DIM  128
#define N_NODES  10000
#define N_EDGES  640000

typedef __attribute__((ext_vector_type(2))) float v2f;
typedef __attribute__((ext_vector_type(8))) float v8f;

// Workspace layout (float elements)
#define A0_OFF   0
#define A1_OFF   (N_NODES * IN_DIM)
#define D0_OFF   (2 * N_NODES * IN_DIM)
#define D1_OFF   (2 * N_NODES * IN_DIM + N_NODES)
#define WS_FLOATS (2 * N_NODES * IN_DIM + 2 * N_NODES)

__global__ __launch_bounds__(256) void gnn_zero_ws(float* __restrict__ ws) {
    int i = blockIdx.x * blockDim.x + threadIdx.x;
    if (i < WS_FLOATS) ws[i] = 0.0f;
}

// One wave per edge (loop-strided). Lane j handles floats [4j, 4j+4) of the
// 128-float row: float4 load from X[src], 4 fp32 atomics into A_type[dst].
__global__ __launch_bounds__(256) void gnn_edge_scatter(
    const float* __restrict__ X,
    const int*   __restrict__ edge_index,   // [2, N_EDGES]
    const int*   __restrict__ edge_types,   // [N_EDGES]
    float*       __restrict__ ws)
{
    float* A0 = ws + A0_OFF;
    float* A1 = ws + A1_OFF;
    float* D0 = ws + D0_OFF;
    float* D1 = ws + D1_OFF;

    const int lane        = threadIdx.x & 31;
    const int wave        = (blockIdx.x * blockDim.x + threadIdx.x) >> 5;
    const int waves_total = (gridDim.x * blockDim.x) >> 5;

    for (int e = wave; e < N_EDGES; e += waves_total) {
        const int src = edge_index[e];
        const int dst = edge_index[N_EDGES + e];
        const int t   = edge_types[e];

        const float4 xv = ((const float4*)(X + (size_t)src * IN_DIM))[lane];

        float* acc = (t == 0 ? A0 : A1) + (size_t)dst * IN_DIM + lane * 4;
        __hip_atomic_fetch_add(acc + 0, xv.x, __ATOMIC_RELAXED, __HIP_MEMORY_SCOPE_AGENT);
        __hip_atomic_fetch_add(acc + 1, xv.y, __ATOMIC_RELAXED, __HIP_MEMORY_SCOPE_AGENT);
        __hip_atomic_fetch_add(acc + 2, xv.z, __ATOMIC_RELAXED, __HIP_MEMORY_SCOPE_AGENT);
        __hip_atomic_fetch_add(acc + 3, xv.w, __ATOMIC_RELAXED, __HIP_MEMORY_SCOPE_AGENT);

        if (lane == 0) {
            __hip_atomic_fetch_add((t == 0 ? D0 : D1) + dst, 1.0f,
                                   __ATOMIC_RELAXED, __HIP_MEMORY_SCOPE_AGENT);
        }
    }
}

// Fused triple GEMM + bias/degree epilogue + ReLU.
// Block = 256 threads = 8 waves; wave w owns out-channel tile [16w, 16w+16),
// block b owns node tile [16b, 16b+16). K=128 fully unrolled:
// 32 steps x 3 V_WMMA_F32_16X16X4_F32 chained into one v8f accumulator.
__global__ __launch_bounds__(256) void gnn_wmma_gemm(
    const float* __restrict__ X,
    const float* __restrict__ Wd, const float* __restrict__ bd,
    const float* __restrict__ Wc, const float* __restrict__ bc,
    const float* __restrict__ Wsf, const float* __restrict__ bs,
    const float* __restrict__ ws,
    float*       __restrict__ out)
{
    const float* A0 = ws + A0_OFF;
    const float* A1 = ws + A1_OFF;
    const float* D0 = ws + D0_OFF;
    const float* D1 = ws + D1_OFF;

    const int lane = threadIdx.x & 31;
    const int r    = lane & 15;        // row index within tile (M for A, N for B)
    const int h    = lane >> 4;        // half-select: K offset 2h per ISA f32 layout
    const int o_base = (threadIdx.x >> 5) * 16;   // wave -> out-channel tile
    const int m_base = blockIdx.x * 16;           // block -> node tile

    // A-matrix 16x4 f32: lane(0-15)=M, VGPR0/1 = K{0,1}; lanes 16-31 = K{2,3}.
    // B-matrix 4x16 f32: lane(0-15)=N, VGPR0/1 = K{0,1}; lanes 16-31 = K{2,3}.
    // B[k][o] = W[o][k] (W row-major [OUT][IN]) -> same addressing pattern.
    const size_t arow = (size_t)(m_base + r) * IN_DIM + 2 * h;
    const size_t brow = (size_t)(o_base + r) * IN_DIM + 2 * h;

    v8f c = {};
#pragma unroll
    for (int k = 0; k < IN_DIM; k += 4) {
        v2f a0  = *(const v2f*)(A0  + arow + k);
        v2f wdv = *(const v2f*)(Wd  + brow + k);
        c = __builtin_amdgcn_wmma_f32_16x16x4_f32(false, a0, false, wdv,
                                                  (short)0, c, false, false);
        v2f a1  = *(const v2f*)(A1  + arow + k);
        v2f wcv = *(const v2f*)(Wc  + brow + k);
        c = __builtin_amdgcn_wmma_f32_16x16x4_f32(false, a1, false, wcv,
                                                  (short)0, c, false, false);
        v2f ax  = *(const v2f*)(X   + arow + k);
        v2f wsv = *(const v2f*)(Wsf + brow + k);
        c = __builtin_amdgcn_wmma_f32_16x16x4_f32(false, ax, false, wsv,
                                                  (short)0, c, false, false);
    }

    // C/D 16x16 f32 layout: VGPR i, lanes 0-15 -> M=i, lanes 16-31 -> M=8+i; N=lane&15.
    const int   o   = o_base + r;
    const float bdo = bd[o];
    const float bco = bc[o];
    const float bso = bs[o];
#pragma unroll
    for (int i = 0; i < 8; ++i) {
        const int m = m_base + i + 8 * h;
        float v = c[i] + D0[m] * bdo + D1[m] * bco + bso;
        out[(size_t)m * OUT_DIM + o] = v > 0.0f ? v : 0.0f;
    }
}

extern "C" void kernel_launch(void* const* d_in, const int* in_sizes, int n_in,
                              void* d_out, int out_size, void* d_ws, size_t ws_size,
                              hipStream_t stream) {
    const float* X   = (const float*)d_in[0];
    const int*   ei  = (const int*)d_in[1];
    const int*   et  = (const int*)d_in[2];
    const float* Wd  = (const float*)d_in[3];
    const float* bd  = (const float*)d_in[4];
    const float* Wc  = (const float*)d_in[5];
    const float* bc  = (const float*)d_in[6];
    const float* Wsf = (const float*)d_in[7];
    const float* bs  = (const float*)d_in[8];
    float* out = (float*)d_out;
    float* ws  = (float*)d_ws;

    (void)in_sizes; (void)n_in; (void)out_size; (void)ws_size;

    // 1) Zero accumulators (A0, A1, deg0, deg1) in workspace.
    gnn_zero_ws<<<(WS_FLOATS + 255) / 256, 256, 0, stream>>>(ws);

    // 2) Edge scatter: type-split row accumulation + degree counts.
    //    4096 blocks x 8 waves -> ~20 edges per wave.
    gnn_edge_scatter<<<4096, 256, 0, stream>>>(X, ei, et, ws);

    // 3) Fused WMMA GEMMs + epilogue. 625 node tiles x 8 channel-tile waves.
    gnn_wmma_gemm<<<N_NODES / 16, 256, 0, stream>>>(X, Wd, bd, Wc, bc, Wsf, bs,
                                                    ws, out);
}